// Encoder_68693706932594
// MI455X (gfx1250) — compile-verified
//
#include <hip/hip_runtime.h>
#include <hip/hip_bf16.h>

#define FEAT 96
#define LSTRIDE 100   // padded A-tile LDS row stride (words): 400B rows -> 16B-aligned,
                      // 36*m mod 64 distinct for m in [0,16) -> conflict-free frags
#define WPAIR 192     // W pair-row stride in words: 96 cols * 2 interleaved k-values

typedef float v2f __attribute__((ext_vector_type(2)));
typedef float v8f __attribute__((ext_vector_type(8)));

// ---------------------------------------------------------------------------
// Utility: zero a float buffer
// ---------------------------------------------------------------------------
__global__ void zero_kernel(float* __restrict__ p, long long n) {
    long long i = (long long)blockIdx.x * blockDim.x + threadIdx.x;
    if (i < n) p[i] = 0.0f;
}

// deg[dst[e]] += 1
__global__ void degree_kernel(const int* __restrict__ dst, float* __restrict__ deg, int E) {
    int e = blockIdx.x * blockDim.x + threadIdx.x;
    if (e < E) atomicAdd(&deg[dst[e]], 1.0f);
}

// dinv[i] = deg>0 ? rsqrt(max(deg,1)) : 0
__global__ void dinv_kernel(const float* __restrict__ deg, float* __restrict__ dinv, int N) {
    int i = blockIdx.x * blockDim.x + threadIdx.x;
    if (i < N) {
        float d = deg[i];
        dinv[i] = (d > 0.0f) ? rsqrtf(fmaxf(d, 1.0f)) : 0.0f;
    }
}

// ---------------------------------------------------------------------------
// C[N,96] = A[N,96] @ W[96,96] via V_WMMA_F32_16X16X4_F32 (full f32).
// Block = 128 threads = 4 waves. W staged once per block into LDS in a
// pair-interleaved, fragment-ready layout:
//     Wlds[kp*192 + n*2 + j] = W[2*kp + j][n]
// so each B fragment is ONE aligned ds_load_b64 directly into an even VGPR
// pair (no repacking movs). Each wave stages its 16x96 A tile into LDS with
// coalesced b128; A fragments are contiguous 8B pairs too.
//
// ISA VGPR layouts (CDNA5 ISA):
//   A 16x4 : lanes 0-15 hold M=lane with K={0,1}; lanes 16-31 K={2,3}
//   B 4x16 : lanes 0-15 hold N=lane with K={0,1}; lanes 16-31 K={2,3}
//   C/D    : 8 VGPRs; lanes 0-15 N=lane, VGPR i -> M=i; lanes 16-31 M=i+8
// ---------------------------------------------------------------------------
__global__ void __launch_bounds__(128)
gemm_wmma_f32(const float* __restrict__ A,
              const float* __restrict__ W,
              float* __restrict__ C,
              int nrows) {
    __shared__ float Wlds[(FEAT / 2) * WPAIR];    // 48 * 192 = 9216 words = 36 KB
    __shared__ float Alds[4][16 * LSTRIDE];       // 4 waves * 6.4 KB

    const int tid  = threadIdx.x;
    const int lane = tid & 31;
    const int wv   = tid >> 5;                    // wave in block: 0..3
    const int wid  = blockIdx.x * 4 + wv;
    const int row0 = wid << 4;
    const bool active = (row0 < nrows);           // wave-uniform

    // ---- stage W (96x96) into LDS, pair-interleaved (coalesced b128 reads) ----
    for (int i = tid; i < FEAT * (FEAT / 4); i += 128) {   // 2304 float4
        const int r  = i / (FEAT / 4);
        const int c0 = (i % (FEAT / 4)) * 4;
        const float4 v = *(const float4*)(W + r * FEAT + c0);
        float* __restrict__ dp = &Wlds[(r >> 1) * WPAIR + (r & 1)];
        dp[(c0 + 0) * 2] = v.x;
        dp[(c0 + 1) * 2] = v.y;
        dp[(c0 + 2) * 2] = v.z;
        dp[(c0 + 3) * 2] = v.w;
    }

    // ---- stage this wave's 16x96 A tile into LDS, coalesced float4 ----
    if (active) {
        for (int i = lane; i < 16 * (FEAT / 4); i += 32) { // 384 float4, 12/lane
            const int r  = i / (FEAT / 4);
            const int c4 = i % (FEAT / 4);
            int gr = row0 + r;
            if (gr > nrows - 1) gr = nrows - 1;            // clamp (N%16==0 -> no-op)
            const float4 v = ((const float4*)(A + (long long)gr * FEAT))[c4];
            *(float4*)&Alds[wv][r * LSTRIDE + c4 * 4] = v; // 16B-aligned (400B rows)
        }
    }
    __syncthreads();           // all threads reach this before any early-out
    if (!active) return;

    const int m  = lane & 15;             // M for A-frag / N for B-frag & C
    const int kh = (lane >> 4) << 1;      // K sub-offset within k-step: 0 or 2
    const float* __restrict__ arow = &Alds[wv][m * LSTRIDE];

    v8f acc[6];
#pragma unroll
    for (int t = 0; t < 6; ++t) acc[t] = {};

    for (int k0 = 0; k0 < FEAT; k0 += 4) {
        const v2f a = *(const v2f*)&arow[k0 + kh];           // one ds_load_b64
        const float* __restrict__ wp = &Wlds[((k0 + kh) >> 1) * WPAIR + m * 2];
#pragma unroll
        for (int t = 0; t < 6; ++t) {
            const v2f b = *(const v2f*)&wp[t * 32];          // one ds_load_b64
            acc[t] = __builtin_amdgcn_wmma_f32_16x16x4_f32(
                false, a, false, b, (short)0, acc[t], false, false);
        }
    }

    const int half = lane >> 4;           // 0 -> M 0..7, 1 -> M 8..15
#pragma unroll
    for (int t = 0; t < 6; ++t) {
#pragma unroll
        for (int i = 0; i < 8; ++i) {
            int r = row0 + half * 8 + i;
            if (r < nrows) C[(long long)r * FEAT + t * 16 + m] = acc[t][i];
        }
    }
}

// ---------------------------------------------------------------------------
// Edge aggregation: out[dst[e]] += h[src[e]] * (dinv[src]*dinv[dst])
// One wave per edge; 32 lanes cover the 96 features in 3 coalesced strides.
// Prefetch the gathered row while the dinv loads are in flight.
// ---------------------------------------------------------------------------
__global__ void aggregate_kernel(const float* __restrict__ h,
                                 const int* __restrict__ src,
                                 const int* __restrict__ dst,
                                 const float* __restrict__ dinv,
                                 float* __restrict__ out,
                                 int E) {
    const int lane = threadIdx.x & 31;
    const long long e = (long long)blockIdx.x * (blockDim.x >> 5) + (threadIdx.x >> 5);
    if (e >= E) return;
    const int s = src[e];
    const int d = dst[e];
    const float* __restrict__ hs = h + (long long)s * FEAT;
    __builtin_prefetch(hs + lane, 0, 0);          // global_prefetch_b8 (gather row)
    const float norm = dinv[s] * dinv[d];
    float* __restrict__ od = out + (long long)d * FEAT;
#pragma unroll
    for (int i = 0; i < 3; ++i) {
        const int f = lane + i * 32;
        atomicAdd(&od[f], hs[f] * norm);
    }
}

// hidden = relu(agg + b) in place
__global__ void bias_relu_kernel(float* __restrict__ x, const float* __restrict__ b,
                                 long long n) {
    long long i = (long long)blockIdx.x * blockDim.x + threadIdx.x;
    if (i < n) {
        float v = x[i] + b[(int)(i % FEAT)];
        x[i] = v > 0.0f ? v : 0.0f;
    }
}

// out[0:nf) += b2a, out[nf:2nf) += b2b
__global__ void bias2_kernel(float* __restrict__ out, const float* __restrict__ ba,
                             const float* __restrict__ bb, long long nf) {
    long long i = (long long)blockIdx.x * blockDim.x + threadIdx.x;
    if (i < 2 * nf) {
        const float* b = (i < nf) ? ba : bb;
        out[i] += b[(int)(i % FEAT)];
    }
}

// ---------------------------------------------------------------------------
extern "C" void kernel_launch(void* const* d_in, const int* in_sizes, int n_in,
                              void* d_out, int out_size, void* d_ws, size_t ws_size,
                              hipStream_t stream) {
    const float* x   = (const float*)d_in[0];
    const float* W1  = (const float*)d_in[1];
    const float* b1  = (const float*)d_in[2];
    const float* W2a = (const float*)d_in[3];
    const float* b2a = (const float*)d_in[4];
    const float* W2b = (const float*)d_in[5];
    const float* b2b = (const float*)d_in[6];
    const int*   ei  = (const int*)d_in[7];

    const int N = in_sizes[0] / FEAT;     // 50000
    const int E = in_sizes[7] / 2;        // 899999
    const int* src = ei;
    const int* dst = ei + E;

    const long long NF = (long long)N * FEAT;

    float* ws   = (float*)d_ws;
    float* deg  = ws;                     // N
    float* dinv = ws + N;                 // N
    float* h    = ws + 2LL * N;           // N*FEAT (GEMM output / messages)
    float* agg  = h + NF;                 // N*FEAT (aggregation, then hidden1)

    float* out  = (float*)d_out;          // [mu (NF) | logstd (NF)]

    const int T = 256;
    const int wavesPerBlk = T / 32;
    const int gemmWaves   = (N + 15) / 16;
    const int gemmBlocks  = (gemmWaves + 3) / 4;           // 128 threads/block
    const int aggBlocks   = (E + wavesPerBlk - 1) / wavesPerBlk;

    // ---- normalization ----
    zero_kernel<<<(int)((N + T - 1) / T), T, 0, stream>>>(deg, N);
    zero_kernel<<<(int)((NF + T - 1) / T), T, 0, stream>>>(agg, NF);
    zero_kernel<<<(int)((2 * NF + T - 1) / T), T, 0, stream>>>(out, 2 * NF);
    degree_kernel<<<(E + T - 1) / T, T, 0, stream>>>(dst, deg, E);
    dinv_kernel<<<(N + T - 1) / T, T, 0, stream>>>(deg, dinv, N);

    // ---- layer 1: hidden1 = relu(scatter(norm * (x@W1)[src]) + b1) ----
    gemm_wmma_f32<<<gemmBlocks, 128, 0, stream>>>(x, W1, h, N);
    aggregate_kernel<<<aggBlocks, T, 0, stream>>>(h, src, dst, dinv, agg, E);
    bias_relu_kernel<<<(int)((NF + T - 1) / T), T, 0, stream>>>(agg, b1, NF);

    // ---- layer 2a: mu ----
    gemm_wmma_f32<<<gemmBlocks, 128, 0, stream>>>(agg, W2a, h, N);
    aggregate_kernel<<<aggBlocks, T, 0, stream>>>(h, src, dst, dinv, out, E);

    // ---- layer 2b: logstd ----
    gemm_wmma_f32<<<gemmBlocks, 128, 0, stream>>>(agg, W2b, h, N);
    aggregate_kernel<<<aggBlocks, T, 0, stream>>>(h, src, dst, dinv, out + NF, E);

    // ---- output bias ----
    bias2_kernel<<<(int)((2 * NF + T - 1) / T), T, 0, stream>>>(out, b2a, b2b, NF);
}